// ImpalaLoss_85590108275284
// MI455X (gfx1250) — compile-verified
//
#include <hip/hip_runtime.h>
#include <cstdint>

#define B_   2048
#define T_   512
#define A_   18
#define TP   510   // T-2
#define TW   511   // T-1
#define TT   28    // scan tile width (time)
#define NBROW 128  // scan rows per block

typedef float v2f __attribute__((ext_vector_type(2)));
typedef float v8f __attribute__((ext_vector_type(8)));
typedef unsigned int v4u __attribute__((ext_vector_type(4)));
typedef int v4i __attribute__((ext_vector_type(4)));
typedef int v8i __attribute__((ext_vector_type(8)));

// ---------------------------------------------------------------------------
// Wave-level f32 sum using V_WMMA_F32_16X16X4_F32 with B = ones.
// A (16x4 f32, 2 VGPRs/lane): lanes 0-15 -> A[m][0],A[m][1]; lanes 16-31 -> A[m][2],A[m][3].
// With a[0]=x, a[1]=0:  rowsum(m) = x_m + x_{m+16}.  D[m][n] = rowsum(m).
// Lane n<16 holds D[0..7][n], lane n>=16 holds D[8..15][n-16] in 8 VGPRs.
// total = (sum of 8 D comps in lane 0) + (sum of 8 D comps in lane 16).
// Pure f32, deterministic.
// ---------------------------------------------------------------------------
__device__ __forceinline__ float wave_sum32(float x) {
#if __has_builtin(__builtin_amdgcn_wmma_f32_16x16x4_f32)
  v2f a; a[0] = x;    a[1] = 0.0f;
  v2f b; b[0] = 1.0f; b[1] = 1.0f;   // all-ones B regardless of exact striping
  v8f c = {};
  c = __builtin_amdgcn_wmma_f32_16x16x4_f32(false, a, false, b, (short)0, c,
                                            false, false);
  float s = c[0] + c[1] + c[2] + c[3] + c[4] + c[5] + c[6] + c[7];
  return __shfl(s, 0, 32) + __shfl(s, 16, 32);
#else
  for (int off = 16; off; off >>= 1) x += __shfl_xor(x, off, 32);
  return x;
#endif
}

// Block (256 threads = 8 waves) sum; result valid in thread 0.
__device__ __forceinline__ float block_sum(float x, float* red8) {
  float w = wave_sum32(x);
  int wid = threadIdx.x >> 5;
  if ((threadIdx.x & 31) == 0) red8[wid] = w;
  __syncthreads();
  float r = 0.0f;
  if (threadIdx.x == 0) {
    int nw = (blockDim.x + 31) >> 5;
    for (int i = 0; i < nw; ++i) r += red8[i];
  }
  __syncthreads();
  return r;
}

// ---------------------------------------------------------------------------
// TDM: 1-D tile load, global -> LDS.  data_size=4B, tile_dim0=nelem,
// tensor_dim0 huge (no OOB), type=2, count=1.  Per-wave issue; caller waits
// s_wait_tensorcnt then __syncthreads().
// ---------------------------------------------------------------------------
__device__ __forceinline__ void tdm_load_1d(uint32_t lds_off, uint64_t gaddr,
                                            uint32_t nelem) {
  v4u g0;
  g0[0] = 1u;                                   // count=1, user desc
  g0[1] = lds_off;                              // lds_addr (bytes)
  g0[2] = (uint32_t)(gaddr & 0xFFFFFFFFu);      // global_addr[31:0]
  g0[3] = (uint32_t)(gaddr >> 32) | 0x80000000u;// global_addr[56:32] | type=2
  const uint32_t dim0 = 0x10000000u;            // tensor_dim0: 268M elems (no OOB)
  v8i g1;
  g1[0] = (int)(2u << 16);                      // data_size=2 (4 bytes)
  g1[1] = (int)((dim0 & 0xFFFFu) << 16);        // tensor_dim0[15:0]
  g1[2] = (int)(((dim0 >> 16) & 0xFFFFu) | (1u << 16)); // dim0 hi | tensor_dim1=1
  g1[3] = (int)(nelem << 16);                   // tile_dim0
  g1[4] = 0;                                    // tile_dim1=0, tile_dim2=0
  g1[5] = (int)nelem;                           // dim0_stride lo (unused for 1D)
  g1[6] = 0;
  g1[7] = 0;
  v4i z4 = {0, 0, 0, 0};
#if __clang_major__ >= 23
  v8i z8 = {0, 0, 0, 0, 0, 0, 0, 0};
  __builtin_amdgcn_tensor_load_to_lds(g0, g1, z4, z4, z8, 0);
#else
  __builtin_amdgcn_tensor_load_to_lds(g0, g1, z4, z4, 0);
#endif
}

// ---------------------------------------------------------------------------
// K1: fused gather + elementwise + entropy.  One thread per global row
// r = b*512 + t.  pi rows for the block are TDM-staged into LDS (contiguous
// 18 KB), read once from HBM.  All outputs batch-major, coalesced.
// ---------------------------------------------------------------------------
__global__ void __launch_bounds__(256)
k1_fused(const float* __restrict__ pi, const float* __restrict__ mu,
         const int* __restrict__ action, const float* __restrict__ values,
         const float* __restrict__ next_values, const float* __restrict__ reward,
         const float* __restrict__ done,
         float* __restrict__ csB, float* __restrict__ dctB,
         float* __restrict__ deltaAB, float* __restrict__ dB1,
         float* __restrict__ dB2, float* __restrict__ lpB,
         float* __restrict__ entP) {
  __shared__ float tile[256 * A_];   // 4608 f32 = 18 KB
  __shared__ float red8[8];

  const int tid = threadIdx.x;
  const int r0  = blockIdx.x * 256;
  const int r   = r0 + tid;

  // Each of the 8 waves TDM-loads its 576-float slice of this block's pi rows.
  {
    int uwid = __builtin_amdgcn_readfirstlane(tid >> 5);
    const uint32_t wslice = (256 * A_) / 8;                  // 576 floats
    uint32_t lds_base = (uint32_t)(uintptr_t)(&tile[0]);
    uint64_t gbase = (uint64_t)(uintptr_t)pi + (uint64_t)r0 * (A_ * 4);
    tdm_load_1d(lds_base + (uint32_t)uwid * wslice * 4u,
                gbase + (uint64_t)uwid * wslice * 4u, wslice);
    __builtin_amdgcn_s_wait_tensorcnt(0);
  }
  __syncthreads();

  const int b = r >> 9;        // r / 512
  const int t = r & (T_ - 1);  // r % 512

  // Entropy over first window: sum pi*log(pi), t < 510.
  float ent = 0.0f;
  if (t < TP) {
    const float* row = &tile[tid * A_];
#pragma unroll
    for (int a = 0; a < A_; ++a) {
      float p = row[a];
      ent += p * logf(p);
    }
  }

  if (t < TW) {
    int   a  = action[r];
    float p  = tile[tid * A_ + a];
    float m  = mu[(size_t)r * A_ + a];
    float lp = logf(p + 1e-8f);
    float lm = logf(m + 1e-8f);
    float rho  = expf(lp - lm);
    float c    = fminf(rho, 1.0f);               // clipped rho == cs (thresholds 1.0)
    float disc = (1.0f - done[r]) * 0.99f;
    float vt   = values[r];
    float vtp1 = values[r + 1];                  // same b: t <= 510 < 511
    size_t o = (size_t)b * TW + t;
    csB[o]     = c;
    dctB[o]    = disc * c;
    deltaAB[o] = c * (reward[r] + disc * vtp1 - vt);
    if (t < TP) lpB[(size_t)b * TP + t] = lp;
    if (t == TP - 1) dB1[b] = c * (reward[r] + disc * next_values[r] - vt);
    if (t == TW - 1) dB2[b] = c * (reward[r] + disc * next_values[r] - vt);
  }

  float bs = block_sum(ent, red8);
  if (tid == 0) entP[blockIdx.x] = bs;
}

// ---------------------------------------------------------------------------
// K2: tiled backward scan.  Thread = one batch row, both windows at once.
// acc = delta_t + (disc*c)_t * acc, t = 509..0.  LDS tiles keep all global
// traffic coalesced (batch-major rows, contiguous time columns).
// ---------------------------------------------------------------------------
__global__ void __launch_bounds__(NBROW)
k2_scan(const float* __restrict__ deltaAB, const float* __restrict__ dctB,
        const float* __restrict__ dB1, const float* __restrict__ dB2,
        float* __restrict__ vmv1, float* __restrict__ vmv2) {
  __shared__ float sD[NBROW * (TT + 2)];   // delta, cols 0..len (stride 30)
  __shared__ float sC[NBROW * (TT + 2)];   // disc*c
  __shared__ float sV1[NBROW * (TT + 1)];  // acc window1 (stride 29)
  __shared__ float sV2[NBROW * (TT + 1)];

  const int tid = threadIdx.x;
  const int b0  = blockIdx.x * NBROW;
  const float d1last = dB1[b0 + tid];
  const float d2last = dB2[b0 + tid];

  float acc1 = 0.0f, acc2 = 0.0f;
  const int ntiles = (TP + TT - 1) / TT;   // 19 (last tile len=6)

  for (int k = ntiles - 1; k >= 0; --k) {
    const int jb    = k * TT;
    const int len   = min(TT, TP - jb);
    const int ncols = len + 1;             // +1 col for window2 (t = j+1)

    for (int i = tid; i < NBROW * ncols; i += NBROW) {
      int rr = i / ncols, cc = i % ncols;
      size_t g = (size_t)(b0 + rr) * TW + jb + cc;   // jb+cc <= 510 < 511
      sD[rr * (TT + 2) + cc] = deltaAB[g];
      sC[rr * (TT + 2) + cc] = dctB[g];
    }
    __syncthreads();

    for (int jl = len - 1; jl >= 0; --jl) {
      int j = jb + jl;
      float d1, d2;
      if (j == TP - 1) { d1 = d1last; d2 = d2last; }   // bootstrap deltas
      else {
        d1 = sD[tid * (TT + 2) + jl];
        d2 = sD[tid * (TT + 2) + jl + 1];
      }
      float c1 = sC[tid * (TT + 2) + jl];
      float c2 = sC[tid * (TT + 2) + jl + 1];
      acc1 = d1 + c1 * acc1;
      acc2 = d2 + c2 * acc2;
      sV1[tid * (TT + 1) + jl] = acc1;
      sV2[tid * (TT + 1) + jl] = acc2;
    }
    __syncthreads();

    for (int i = tid; i < NBROW * len; i += NBROW) {
      int rr = i / len, cc = i % len;
      size_t g = (size_t)(b0 + rr) * TP + jb + cc;
      vmv1[g] = sV1[rr * (TT + 1) + cc];
      vmv2[g] = sV2[rr * (TT + 1) + cc];
    }
    __syncthreads();
  }
}

// ---------------------------------------------------------------------------
// K3: pg-advantage, pi_loss and value_loss partials.  One thread per (b, j).
// value_loss term: (vs - v)^2 == (vs_minus_v)^2 == vmv1^2.
// ---------------------------------------------------------------------------
__global__ void __launch_bounds__(256)
k3_loss(const float* __restrict__ values, const float* __restrict__ reward,
        const float* __restrict__ done, const float* __restrict__ csB,
        const float* __restrict__ lpB, const float* __restrict__ vmv1,
        const float* __restrict__ vmv2,
        float* __restrict__ piP, float* __restrict__ vlP) {
  __shared__ float red8[8];
  const int tid = threadIdx.x;
  const int e   = blockIdx.x * 256 + tid;     // exact: B*TP == 4080*256

  const int b = e / TP;
  const int j = e - b * TP;
  const size_t rij = (size_t)b * T_ + j;

  float disc  = (1.0f - done[rij]) * 0.99f;
  float vs_p1 = vmv2[e] + values[rij + 1];    // vs_plus_1 = acc2 + second_v
  float adv   = csB[(size_t)b * TW + j] *
                (reward[rij] + disc * vs_p1 - values[rij]);
  float pip = -lpB[e] * adv;
  float a1  = vmv1[e];
  float vlp = a1 * a1;

  float s1 = block_sum(pip, red8);
  float s2 = block_sum(vlp, red8);
  if (tid == 0) { piP[blockIdx.x] = s1; vlP[blockIdx.x] = s2; }
}

// ---------------------------------------------------------------------------
// K4: final deterministic reduction of the partial arrays -> 4 scalars.
// ---------------------------------------------------------------------------
__global__ void __launch_bounds__(256)
k4_final(const float* __restrict__ entP, int nent,
         const float* __restrict__ piP, const float* __restrict__ vlP, int nblk,
         float* __restrict__ out) {
  __shared__ float red8[8];
  const int tid = threadIdx.x;
  float se = 0.0f, sp = 0.0f, sv = 0.0f;
  for (int i = tid; i < nent; i += 256) se += entP[i];
  for (int i = tid; i < nblk; i += 256) { sp += piP[i]; sv += vlP[i]; }
  float ent     = block_sum(se, red8);
  float pi_loss = block_sum(sp, red8);
  float vl      = block_sum(sv, red8);
  if (tid == 0) {
    out[0] = pi_loss + vl * 1.0f + ent + 0.05f;  // tot (BASELINE=1, ENT=0.05)
    out[1] = pi_loss;
    out[2] = vl;
    out[3] = ent;
  }
}

// ---------------------------------------------------------------------------
extern "C" void kernel_launch(void* const* d_in, const int* in_sizes, int n_in,
                              void* d_out, int out_size, void* d_ws,
                              size_t ws_size, hipStream_t stream) {
  const float* pi          = (const float*)d_in[0];
  const float* mu          = (const float*)d_in[1];
  const int*   action      = (const int*)d_in[2];
  const float* values      = (const float*)d_in[3];
  const float* next_values = (const float*)d_in[4];
  const float* reward      = (const float*)d_in[5];
  const float* done        = (const float*)d_in[6];
  float* out = (float*)d_out;

  float* ws      = (float*)d_ws;
  float* csB     = ws;                               // B*TW
  float* dctB    = csB     + (size_t)B_ * TW;        // B*TW
  float* deltaAB = dctB    + (size_t)B_ * TW;        // B*TW
  float* dB1     = deltaAB + (size_t)B_ * TW;        // B
  float* dB2     = dB1     + B_;                     // B
  float* lpB     = dB2     + B_;                     // B*TP
  float* vmv1    = lpB     + (size_t)B_ * TP;        // B*TP
  float* vmv2    = vmv1    + (size_t)B_ * TP;        // B*TP
  float* entP    = vmv2    + (size_t)B_ * TP;        // 4096
  float* piP     = entP    + 4096;                   // 4080
  float* vlP     = piP     + 4080;                   // 4080

  const int nblk1 = (B_ * T_) / 256;   // 4096
  const int nblk3 = (B_ * TP) / 256;   // 4080

  k1_fused<<<nblk1, 256, 0, stream>>>(pi, mu, action, values, next_values,
                                      reward, done, csB, dctB, deltaAB, dB1,
                                      dB2, lpB, entP);
  k2_scan<<<B_ / NBROW, NBROW, 0, stream>>>(deltaAB, dctB, dB1, dB2, vmv1, vmv2);
  k3_loss<<<nblk3, 256, 0, stream>>>(values, reward, done, csB, lpB, vmv1,
                                     vmv2, piP, vlP);
  k4_final<<<1, 256, 0, stream>>>(entP, nblk1, piP, vlP, nblk3, out);
}